// E2formerCluster_15367392985802
// MI455X (gfx1250) — compile-verified
//
#include <hip/hip_runtime.h>
#include <hip/hip_bf16.h>
#include <stdint.h>

// ---- problem constants (from reference) ----
#define NM_    9      // (LMAX+1)^2
#define NH_    8
#define DH_    32
#define AWD_   32
#define HID_   256
#define EMB_   128
#define KN_    16     // neighbors per node
#define N_NODE 4096
#define M_CLU  512
#define ECH_   416    // AWD + EMB + HID
#define R3_    192    // 2*AWD*(LMAX+1)

typedef float v2f __attribute__((ext_vector_type(2)));
typedef float v8f __attribute__((ext_vector_type(8)));
typedef uint32_t u32x4 __attribute__((ext_vector_type(4)));
typedef uint32_t u32x8 __attribute__((ext_vector_type(8)));

#define V8F_ZERO {0.f,0.f,0.f,0.f,0.f,0.f,0.f,0.f}

// D = A(16x4 f32) * B(4x16 f32) + C ; full-precision CDNA5 WMMA
static __device__ __forceinline__ v8f wmma_f32_k4(v8f acc, float a0, float a1,
                                                  float b0, float b1) {
  v2f A; A[0] = a0; A[1] = a1;
  v2f B; B[0] = b0; B[1] = b1;
  return __builtin_amdgcn_wmma_f32_16x16x4_f32(false, A, false, B, (short)0,
                                               acc, false, false);
}

static __device__ __forceinline__ float sigmoidf_(float x) {
  return 1.0f / (1.0f + __expf(-x));
}

// LayerNorm(16) + SiLU over one row of 16 values (in place)
static __device__ __forceinline__ void ln_silu16(float* rw, const float* g,
                                                 const float* bt) {
  float m = 0.f;
  for (int j = 0; j < 16; ++j) m += rw[j];
  m *= (1.f / 16.f);
  float v = 0.f;
  for (int j = 0; j < 16; ++j) { float d = rw[j] - m; v += d * d; }
  v *= (1.f / 16.f);
  float inv = rsqrtf(v + 1e-5f);
  for (int j = 0; j < 16; ++j) {
    float x = (rw[j] - m) * inv * g[j] + bt[j];
    rw[j] = x * sigmoidf_(x);   // SiLU
  }
}

// ---------------------------------------------------------------------
// TDM helpers (descriptor bit layout per CDNA5 ISA)
// group1 dword0: data_size=4B (code 2), pad_enable, pad_interval=256 DW
// (code 7), pad_amount=2 DW (code 1)  -> LDS row stride 258 floats
#define TDM_G1_DW0  ((2u << 16) | (1u << 20) | (7u << 22) | (1u << 25))

// 2D tile load: tile_dim0 x tile_dim1 elements, row stride strideElems,
// into LDS at ldsAddr with +2DW pad per 256DW -> lda 258.
static __device__ __forceinline__ void tdm_load_tile_2d(
    uint32_t ldsAddr, const void* gptr, uint32_t dim0, uint32_t rows,
    uint32_t strideElems) {
  uint64_t ga = (uint64_t)(uintptr_t)gptr;
  u32x4 g0;
  g0[0] = 1u;                                  // count=1, user mode
  g0[1] = ldsAddr;                             // LDS byte address
  g0[2] = (uint32_t)ga;                        // global addr lo
  g0[3] = (uint32_t)((ga >> 32) & 0x1ffffffu) | (2u << 30);  // hi | type=2
  u32x8 g1;
  g1[0] = TDM_G1_DW0;                          // mask=0, 4B, pad 256DW+2DW
  g1[1] = (dim0 & 0xffffu) << 16;              // tensor_dim0 lo16
  g1[2] = (dim0 >> 16) | ((rows & 0xffffu) << 16);   // dim0 hi | dim1 lo
  g1[3] = (rows >> 16) | ((dim0 & 0xffffu) << 16);   // dim1 hi | tile_dim0
  g1[4] = rows & 0xffffu;                      // tile_dim1 (tile_dim2=0)
  g1[5] = strideElems;                         // tensor_dim0_stride lo32
  g1[6] = 0u;
  g1[7] = 0u;
  asm volatile("tensor_load_to_lds %0, %1" :: "s"(g0), "s"(g1) : "memory");
}

// Gather-mode load: 16 rows selected by 16-bit indices, each row = dim0
// elements at base + idx*strideElems, packed into LDS with lda-258 pad.
// Index dwords are readfirstlane'd so the D# groups are SGPR-allocatable.
static __device__ __forceinline__ void tdm_gather_rows16(
    uint32_t ldsAddr, const void* gbase, uint32_t dim0, uint32_t nrowsTensor,
    uint32_t strideElems, const int* idx /*16 values, same in all lanes*/) {
  uint64_t ga = (uint64_t)(uintptr_t)gbase;
  u32x4 g0;
  g0[0] = 1u | (1u << 31);                     // count=1, gather_mode=1, 16b idx
  g0[1] = ldsAddr;
  g0[2] = (uint32_t)ga;
  g0[3] = (uint32_t)((ga >> 32) & 0x1ffffffu) | (2u << 30);
  u32x8 g1;
  g1[0] = TDM_G1_DW0;
  g1[1] = (dim0 & 0xffffu) << 16;
  g1[2] = (dim0 >> 16) | ((nrowsTensor & 0xffffu) << 16);
  g1[3] = (nrowsTensor >> 16) | ((dim0 & 0xffffu) << 16);
  g1[4] = 16u;                                 // tile_dim1 = #valid indices
  g1[5] = strideElems;
  g1[6] = 0u;
  g1[7] = 0u;
  u32x4 g2, g3;
  for (int j = 0; j < 4; ++j) {
    uint32_t a = ((uint32_t)idx[2 * j] & 0xffffu) |
                 (((uint32_t)idx[2 * j + 1] & 0xffffu) << 16);
    uint32_t b = ((uint32_t)idx[8 + 2 * j] & 0xffffu) |
                 (((uint32_t)idx[9 + 2 * j] & 0xffffu) << 16);
    g2[j] = (uint32_t)__builtin_amdgcn_readfirstlane((int)a);
    g3[j] = (uint32_t)__builtin_amdgcn_readfirstlane((int)b);
  }
  asm volatile("tensor_load_to_lds %0, %1, %2, %3"
               :: "s"(g0), "s"(g1), "s"(g2), "s"(g3) : "memory");
}

// =====================================================================
// Kernel 1: node_dot / key_dot  (SO3 linear with dot_w, per-m 16-row tiles)
//   o[row, d] = sum_c irreps[row, m, c] * dot_w[l(m)][d][c]   (+dot_b if m==0)
// A tile staged by the Tensor Data Mover (pad -> lda 258).
// =====================================================================
__global__ __launch_bounds__(64) void dot_kernel(
    const float* __restrict__ node_irreps, const float* __restrict__ clu_irreps,
    const float* __restrict__ dot_w, const float* __restrict__ dot_b,
    float* __restrict__ node_dot, float* __restrict__ key_dot) {
  __shared__ float sA[16 * 258];        // 16 rows x HID (TDM-padded lda=258)
  __shared__ float sBt[HID_ * AWD_];    // B transposed: [c][d]

  int blk = blockIdx.x;
  const int nodeGroups = (N_NODE / 16) * NM_;
  const float* src; float* dst; int g, m;
  if (blk < nodeGroups) { src = node_irreps; dst = node_dot; g = blk / NM_; m = blk % NM_; }
  else { blk -= nodeGroups; src = clu_irreps; dst = key_dot; g = blk / NM_; m = blk % NM_; }
  const int l = (m == 0) ? 0 : (m < 4 ? 1 : 2);
  const int row0 = g * 16;

  // async DMA of the A tile (16 rows x 256 f32, row stride 2304 elems)
  if ((threadIdx.x >> 5) == 0) {
    tdm_load_tile_2d((uint32_t)(uintptr_t)&sA[0],
                     src + (size_t)row0 * NM_ * HID_ + (size_t)m * HID_,
                     HID_, 16, NM_ * HID_);
  }
  // meanwhile stage B transposed with the VALU path
  for (int i = threadIdx.x; i < AWD_ * HID_; i += 64) {
    int d = i / HID_, c = i % HID_;
    sBt[c * AWD_ + d] = dot_w[l * AWD_ * HID_ + i];
  }
  if ((threadIdx.x >> 5) == 0) __builtin_amdgcn_s_wait_tensorcnt(0);
  __syncthreads();

  const int wave = threadIdx.x >> 5;      // 0..1 -> 16-col tile
  const int lane = threadIdx.x & 31;
  const int hf = lane >> 4, lr = lane & 15;
  const int colbase = wave * 16;

  v8f acc = V8F_ZERO;
  for (int k = 0; k < HID_; k += 4) {
    float a0 = sA[lr * 258 + k + 2 * hf];
    float a1 = sA[lr * 258 + k + 2 * hf + 1];
    float b0 = sBt[(k + 2 * hf) * AWD_ + colbase + lr];
    float b1 = sBt[(k + 2 * hf + 1) * AWD_ + colbase + lr];
    acc = wmma_f32_k4(acc, a0, a1, b0, b1);
  }
  const float bias = (m == 0) ? dot_b[colbase + lr] : 0.f;   // hoisted
  for (int j = 0; j < 8; ++j) {
    int row = j + 8 * hf;
    dst[(size_t)(row0 + row) * NM_ * AWD_ + m * AWD_ + colbase + lr] = acc[j] + bias;
  }
}

// =====================================================================
// Kernel 2: fused per-node edge pipeline (one block per node)
// =====================================================================
struct MegaParams {
  const float* edge_dis;       // (N,K)
  const float* edge_vec;       // (N,K,3)
  const float* attn_weight;    // (N,K,AWD)
  const int*   atomic_numbers; // (N)
  const unsigned char* attn_mask; // (N,K) bool (assumed 1 byte)
  const int*   sparse_idx;     // (N,K)
  const float* clu;            // (M,NM,HID)
  const float* tgt_emb;        // (256,EMB)
  const float *m0w1, *m0b1, *m0g1, *m0t1;
  const float *m0w2, *m0b2, *m0g2, *m0t2;
  const float *m0w3, *m0b3;
  const float *hw1, *hb1, *hg1, *ht1;
  const float *hw2, *hb2, *hg2, *ht2;
  const float *hw3, *hb3;
  const float *fcw, *fcb, *lng, *lnb, *adot;
  const float *node_dot, *key_dot;
  float* attn_out;             // (N,NM,HID) pre-projection
};

// LDS layout (floats); padded row strides to dodge 64-bank conflicts
#define XLDA      162
#define OFF_XEDGE 0           // 16*162 = 2592  [aw|tgt] part of x_edge
#define OFF_CLUT  2592        // 16*258 = 4128  TDM-gathered cluster rows
#define OFF_R1M0  6720        // 16*18 = 288
#define OFF_R1HD  7008        // 288
#define OFF_R2M0  7296        // 288
#define OFF_R2HD  7584        // 288 -> 7872
#define OFF_SH    7872        // 16*9 = 144
#define OFF_NDOT  8016        // 9*32 = 288
#define OFF_LOGIT 8304        // 16*8 = 128
#define OFF_ALPHA 8432        // 128 -> 8560
#define OFF_RAD   8560        // 16*192 = 3072 -> 11632
#define OFF_VSC   11632       // 16*256 = 4096 -> 15728
#define TLDA      194
#define OFF_X0    0           // 16*194 = 3104  (reuses x_edge region, P3+)
#define OFF_APRE  3104        // 16*256 = 4096 -> 7200 (< 7872, clu/r1 dead)
#define SMEM_F    15728       // ~62.9 KB

__global__ __launch_bounds__(256) void mega_kernel(MegaParams p) {
  __shared__ float sm[SMEM_F];
  __shared__ int s_idx[KN_];
  __shared__ int s_msk[KN_];

  const int n = blockIdx.x;
  const int tid = threadIdx.x;
  const int wave = tid >> 5;
  const int lane = tid & 31;
  const int hf = lane >> 4;
  const int lr = lane & 15;

  // ---------------- P0a: indices, mask, spherical harmonics, node_dot stage
  if (tid < KN_) {
    int idx = p.sparse_idx[n * KN_ + tid];
    s_idx[tid] = idx;
    s_msk[tid] = p.attn_mask[n * KN_ + tid] ? 1 : 0;
    float dis = p.edge_dis[n * KN_ + tid];
    float inv = 1.0f / fmaxf(dis, 1e-9f);
    float x = p.edge_vec[(n * KN_ + tid) * 3 + 0] * inv;
    float y = p.edge_vec[(n * KN_ + tid) * 3 + 1] * inv;
    float z = p.edge_vec[(n * KN_ + tid) * 3 + 2] * inv;
    float* sh = sm + OFF_SH + tid * NM_;
    sh[0] = 0.28209479177387814f;
    sh[1] = 0.4886025119029199f * y;
    sh[2] = 0.4886025119029199f * z;
    sh[3] = 0.4886025119029199f * x;
    sh[4] = 1.0925484305920792f * x * y;
    sh[5] = 1.0925484305920792f * y * z;
    sh[6] = 0.31539156525252005f * (3.0f * z * z - 1.0f);
    sh[7] = 1.0925484305920792f * x * z;
    sh[8] = 0.5462742152960396f * (x * x - y * y);
  }
  for (int i = tid; i < NM_ * AWD_; i += 256)
    sm[OFF_NDOT + i] = p.node_dot[(size_t)n * NM_ * AWD_ + i];
  if (tid == 0) {                 // warm L2 for streamed weights
    __builtin_prefetch(p.fcw, 0, 3);
    __builtin_prefetch(p.hw3, 0, 3);
  }
  __syncthreads();

  // ---------------- P0b: TDM gather of the 16 neighbor cluster l0-rows
  // (row k = clu[s_idx[k], 0, :256], landing at lda 258), overlapped with
  // the VALU fill of the [aw|tgt] part of x_edge.
  if (wave == 0) {
    tdm_gather_rows16((uint32_t)(uintptr_t)&sm[OFF_CLUT], p.clu,
                      HID_, M_CLU, NM_ * HID_, s_idx);
  }
  const int zat = p.atomic_numbers[n];
  for (int i = tid; i < KN_ * (AWD_ + EMB_); i += 256) {
    int k = i / (AWD_ + EMB_), c = i % (AWD_ + EMB_);
    float v;
    if (c < AWD_)
      v = s_msk[k] ? 0.0f : p.attn_weight[(n * KN_ + k) * AWD_ + c];
    else
      v = p.tgt_emb[zat * EMB_ + (c - AWD_)];
    sm[OFF_XEDGE + k * XLDA + c] = v;
  }
  if (wave == 0) __builtin_amdgcn_s_wait_tensorcnt(0);
  __syncthreads();

  // ---------------- P1: radial MLP layer 1 (two 16x416x16 WMMA GEMMs)
  // A is split across the [aw|tgt] tile (k<160) and the TDM tile (k>=160).
  if (wave < 2) {
    const float* W = (wave == 0) ? p.m0w1 : p.hw1;
    const float* B = (wave == 0) ? p.m0b1 : p.hb1;
    float* dst = sm + ((wave == 0) ? OFF_R1M0 : OFF_R1HD);
    v8f acc = V8F_ZERO;
    for (int k = 0; k < AWD_ + EMB_; k += 4) {
      float a0 = sm[OFF_XEDGE + lr * XLDA + k + 2 * hf];
      float a1 = sm[OFF_XEDGE + lr * XLDA + k + 2 * hf + 1];
      float b0 = W[(k + 2 * hf) * 16 + lr];
      float b1 = W[(k + 2 * hf + 1) * 16 + lr];
      acc = wmma_f32_k4(acc, a0, a1, b0, b1);
    }
    for (int k = AWD_ + EMB_; k < ECH_; k += 4) {
      int kk = k - (AWD_ + EMB_);
      float a0 = sm[OFF_CLUT + lr * 258 + kk + 2 * hf];
      float a1 = sm[OFF_CLUT + lr * 258 + kk + 2 * hf + 1];
      float b0 = W[(k + 2 * hf) * 16 + lr];
      float b1 = W[(k + 2 * hf + 1) * 16 + lr];
      acc = wmma_f32_k4(acc, a0, a1, b0, b1);
    }
    const float bb = B[lr];
    for (int j = 0; j < 8; ++j) dst[(j + 8 * hf) * 18 + lr] = acc[j] + bb;
  }
  __syncthreads();
  if (tid < 32) {
    int which = tid >> 4, row = tid & 15;
    ln_silu16(sm + (which ? OFF_R1HD : OFF_R1M0) + row * 18,
              which ? p.hg1 : p.m0g1, which ? p.ht1 : p.m0t1);
  }
  __syncthreads();

  // ---------------- P2a: layer 2 (16x16x16)
  if (wave < 2) {
    const float* A = sm + ((wave == 0) ? OFF_R1M0 : OFF_R1HD);
    const float* W = (wave == 0) ? p.m0w2 : p.hw2;
    const float* B = (wave == 0) ? p.m0b2 : p.hb2;
    float* dst = sm + ((wave == 0) ? OFF_R2M0 : OFF_R2HD);
    v8f acc = V8F_ZERO;
    for (int k = 0; k < 16; k += 4) {
      float a0 = A[lr * 18 + k + 2 * hf];
      float a1 = A[lr * 18 + k + 2 * hf + 1];
      float b0 = W[(k + 2 * hf) * 16 + lr];
      float b1 = W[(k + 2 * hf + 1) * 16 + lr];
      acc = wmma_f32_k4(acc, a0, a1, b0, b1);
    }
    const float bb = B[lr];
    for (int j = 0; j < 8; ++j) dst[(j + 8 * hf) * 18 + lr] = acc[j] + bb;
  }
  __syncthreads();
  if (tid < 32) {
    int which = tid >> 4, row = tid & 15;
    ln_silu16(sm + (which ? OFF_R2HD : OFF_R2M0) + row * 18,
              which ? p.hg2 : p.m0g2, which ? p.ht2 : p.m0t2);
  }
  __syncthreads();

  // ---------------- P2b: layer 3 -> rad(16x192) & vscale(16x256); 28 tiles
  for (int t = wave; t < 28; t += 8) {
    bool ism0 = (t < 12);
    int ct = ism0 ? t : (t - 12);
    const float* A = sm + (ism0 ? OFF_R2M0 : OFF_R2HD);
    const float* W = ism0 ? p.m0w3 : p.hw3;
    const float* B = ism0 ? p.m0b3 : p.hb3;
    int wid = ism0 ? R3_ : HID_;
    float* dst = sm + (ism0 ? OFF_RAD : OFF_VSC);
    int col = ct * 16 + lr;
    v8f acc = V8F_ZERO;
    for (int k = 0; k < 16; k += 4) {
      float a0 = A[lr * 18 + k + 2 * hf];
      float a1 = A[lr * 18 + k + 2 * hf + 1];
      float b0 = W[(k + 2 * hf) * wid + col];
      float b1 = W[(k + 2 * hf + 1) * wid + col];
      acc = wmma_f32_k4(acc, a0, a1, b0, b1);
    }
    const float bb = B[col];
    for (int j = 0; j < 8; ++j) dst[(j + 8 * hf) * wid + col] = acc[j] + bb;
  }
  __syncthreads();

  // ---------------- P3: t = x0 * rad  (x0 from sh contractions)
  for (int i = tid; i < KN_ * R3_; i += 256) {
    int k = i / R3_, c = i % R3_;
    int l = c >> 6;             // 0..2
    int piece = (c >> 5) & 1;   // 0=node, 1=key
    int d = c & 31;
    int s = l * l, nmm = 2 * l + 1;
    const float* sh = sm + OFF_SH + k * NM_;
    float acc = 0.f;
    if (piece == 0) {
      for (int mm = 0; mm < nmm; ++mm)
        acc += sh[s + mm] * sm[OFF_NDOT + (s + mm) * AWD_ + d];
    } else {
      const float* kd = p.key_dot + (size_t)s_idx[k] * NM_ * AWD_;
      for (int mm = 0; mm < nmm; ++mm)
        acc += sh[s + mm] * kd[(s + mm) * AWD_ + d];
    }
    sm[OFF_X0 + k * TLDA + c] = acc * sm[OFF_RAD + k * R3_ + c];
  }
  __syncthreads();

  // ---------------- P4: alpha_pre = t @ fc_m0_w + b  (16x192x256)
  for (int t = wave; t < 16; t += 8) {
    int col = t * 16 + lr;
    v8f acc = V8F_ZERO;
    for (int k = 0; k < R3_; k += 4) {
      float a0 = sm[OFF_X0 + lr * TLDA + k + 2 * hf];
      float a1 = sm[OFF_X0 + lr * TLDA + k + 2 * hf + 1];
      float b0 = p.fcw[(k + 2 * hf) * HID_ + col];
      float b1 = p.fcw[(k + 2 * hf + 1) * HID_ + col];
      acc = wmma_f32_k4(acc, a0, a1, b0, b1);
    }
    const float bb = p.fcb[col];
    for (int j = 0; j < 8; ++j)
      sm[OFF_APRE + (j + 8 * hf) * HID_ + col] = acc[j] + bb;
  }
  __syncthreads();

  // ---------------- P5: LN(32) + smooth_leaky + alpha_dot -> logits
  if (tid < 128) {
    int k = tid >> 3, h = tid & 7;
    float* a = sm + OFF_APRE + k * HID_ + h * DH_;
    float m = 0.f;
    for (int d = 0; d < DH_; ++d) m += a[d];
    m *= (1.f / 32.f);
    float v = 0.f;
    for (int d = 0; d < DH_; ++d) { float t = a[d] - m; v += t * t; }
    v *= (1.f / 32.f);
    float inv = rsqrtf(v + 1e-5f);
    float dot = 0.f;
    for (int d = 0; d < DH_; ++d) {
      float x = (a[d] - m) * inv * p.lng[d] + p.lnb[d];
      float sl = 0.6f * x + 0.4f * x * (2.f * sigmoidf_(x) - 1.f);
      dot += sl * p.adot[h * DH_ + d];
    }
    sm[OFF_LOGIT + k * NH_ + h] = s_msk[k] ? -1e9f : dot;
  }
  __syncthreads();

  // ---------------- P6: softmax over the K=16 edges, per head
  if (tid < NH_) {
    int h = tid;
    float mx = -1e30f;
    for (int k = 0; k < KN_; ++k) mx = fmaxf(mx, sm[OFF_LOGIT + k * NH_ + h]);
    float ssum = 0.f;
    for (int k = 0; k < KN_; ++k) {
      float e = __expf(sm[OFF_LOGIT + k * NH_ + h] - mx);
      sm[OFF_ALPHA + k * NH_ + h] = e;
      ssum += e;
    }
    float inv = 1.f / ssum;
    for (int k = 0; k < KN_; ++k) sm[OFF_ALPHA + k * NH_ + h] *= inv;
  }
  __syncthreads();

  // ---------------- P7: attn_out[n,m,c] = sum_k alpha[k,h]*vscale[k,c]*clu[idx,m,c]
  {
    int c = tid;                 // 0..255
    int h = c >> 5;
    float acc[NM_];
    for (int m = 0; m < NM_; ++m) acc[m] = 0.f;
    for (int k = 0; k < KN_; ++k) {
      float w = sm[OFF_ALPHA + k * NH_ + h] * sm[OFF_VSC + k * HID_ + c];
      const float* cr = p.clu + (size_t)s_idx[k] * NM_ * HID_ + c;
#pragma unroll
      for (int m = 0; m < NM_; ++m) acc[m] += w * cr[(size_t)m * HID_];
    }
    float* ob = p.attn_out + (size_t)n * NM_ * HID_ + c;
    for (int m = 0; m < NM_; ++m) ob[(size_t)m * HID_] = acc[m];
  }
}

// =====================================================================
// Kernel 3a: transpose proj_w -> wT[l][c][d] for coalesced B fragments
// =====================================================================
__global__ __launch_bounds__(256) void wT_kernel(const float* __restrict__ projw,
                                                 float* __restrict__ wT) {
  int i = blockIdx.x * 256 + threadIdx.x;
  if (i < 3 * HID_ * HID_) {
    int l = i / (HID_ * HID_);
    int rem = i % (HID_ * HID_);
    int c = rem / HID_, d = rem % HID_;
    wT[i] = projw[l * HID_ * HID_ + d * HID_ + c];
  }
}

// =====================================================================
// Kernel 3b: output SO3 projection. 16 nodes/block; per l-group rows pack
// into exact 16-row tiles: l0 -> 1 tile, l1 -> 3, l2 -> 5 (no padding).
// =====================================================================
__global__ __launch_bounds__(256) void proj_kernel(
    const float* __restrict__ attn, const float* __restrict__ wT,
    const float* __restrict__ projb, float* __restrict__ out) {
  const int n0 = blockIdx.x * 16;
  const int tid = threadIdx.x;
  const int wave = tid >> 5;
  const int lane = tid & 31;
  const int hf = lane >> 4, lr = lane & 15;

  for (int t = wave; t < 144; t += 8) {       // 9 row-tiles x 16 col-tiles
    int rowTile = t % 9;
    int colTile = t / 9;
    int l = (rowTile == 0) ? 0 : (rowTile < 4 ? 1 : 2);
    int rt = rowTile - l * l;
    int span = 2 * l + 1;

    int rr = rt * 16 + lr;
    int node = n0 + rr / span;
    int m = l * l + rr % span;
    const float* arow = attn + (size_t)node * NM_ * HID_ + (size_t)m * HID_;
    const float* bmat = wT + l * HID_ * HID_;  // [c][d]
    int col = colTile * 16 + lr;

    v8f acc = V8F_ZERO;
    for (int k = 0; k < HID_; k += 4) {
      float a0 = arow[k + 2 * hf];
      float a1 = arow[k + 2 * hf + 1];
      float b0 = bmat[(k + 2 * hf) * HID_ + col];
      float b1 = bmat[(k + 2 * hf + 1) * HID_ + col];
      acc = wmma_f32_k4(acc, a0, a1, b0, b1);
    }
    float bb = (l == 0) ? projb[col] : 0.f;
    for (int j = 0; j < 8; ++j) {
      int ro = rt * 16 + j + 8 * hf;
      int nodeo = n0 + ro / span;
      int mo = l * l + ro % span;
      out[(size_t)nodeo * NM_ * HID_ + (size_t)mo * HID_ + col] = acc[j] + bb;
    }
  }
}

// =====================================================================
extern "C" void kernel_launch(void* const* d_in, const int* in_sizes, int n_in,
                              void* d_out, int out_size, void* d_ws, size_t ws_size,
                              hipStream_t stream) {
  (void)in_sizes; (void)n_in; (void)out_size; (void)ws_size;

  // setup_inputs() order, params dict/list flattened in insertion order
  const float* node_irreps   = (const float*)d_in[1];
  const float* edge_dis      = (const float*)d_in[2];
  const float* edge_vec      = (const float*)d_in[3];
  const float* attn_weight   = (const float*)d_in[4];
  const int*   atomic_nums   = (const int*)  d_in[5];
  const unsigned char* amask = (const unsigned char*)d_in[6]; // bool
  const int*   sparse_idx    = (const int*)  d_in[7];
  const float* clu           = (const float*)d_in[8];
  const float* tgt_emb       = (const float*)d_in[9];
  const float* dot_w         = (const float*)d_in[10];
  const float* dot_b         = (const float*)d_in[11];
  const float* m0w1 = (const float*)d_in[12]; const float* m0b1 = (const float*)d_in[13];
  const float* m0g1 = (const float*)d_in[14]; const float* m0t1 = (const float*)d_in[15];
  const float* m0w2 = (const float*)d_in[16]; const float* m0b2 = (const float*)d_in[17];
  const float* m0g2 = (const float*)d_in[18]; const float* m0t2 = (const float*)d_in[19];
  const float* m0w3 = (const float*)d_in[20]; const float* m0b3 = (const float*)d_in[21];
  const float* fcw  = (const float*)d_in[24]; const float* fcb  = (const float*)d_in[25];
  const float* lng  = (const float*)d_in[26]; const float* lnb  = (const float*)d_in[27];
  const float* adot = (const float*)d_in[28];
  const float* hw1  = (const float*)d_in[29]; const float* hb1  = (const float*)d_in[30];
  const float* hg1  = (const float*)d_in[31]; const float* ht1  = (const float*)d_in[32];
  const float* hw2  = (const float*)d_in[33]; const float* hb2  = (const float*)d_in[34];
  const float* hg2  = (const float*)d_in[35]; const float* ht2  = (const float*)d_in[36];
  const float* hw3  = (const float*)d_in[37]; const float* hb3  = (const float*)d_in[38];
  const float* projw = (const float*)d_in[41];
  const float* projb = (const float*)d_in[42];

  // workspace carve-up (floats): dots + pre-proj output + transposed proj_w
  float* ws = (float*)d_ws;
  float* node_dot = ws;                                   // 4096*288
  float* key_dot  = ws + (size_t)N_NODE * NM_ * AWD_;     // 512*288
  float* attn_out = key_dot + (size_t)M_CLU * NM_ * AWD_; // 4096*2304
  float* wT       = attn_out + (size_t)N_NODE * NM_ * HID_; // 3*65536

  dot_kernel<<<(N_NODE / 16) * NM_ + (M_CLU / 16) * NM_, 64, 0, stream>>>(
      node_irreps, clu, dot_w, dot_b, node_dot, key_dot);

  wT_kernel<<<(3 * HID_ * HID_ + 255) / 256, 256, 0, stream>>>(projw, wT);

  MegaParams mp;
  mp.edge_dis = edge_dis; mp.edge_vec = edge_vec; mp.attn_weight = attn_weight;
  mp.atomic_numbers = atomic_nums; mp.attn_mask = amask; mp.sparse_idx = sparse_idx;
  mp.clu = clu; mp.tgt_emb = tgt_emb;
  mp.m0w1 = m0w1; mp.m0b1 = m0b1; mp.m0g1 = m0g1; mp.m0t1 = m0t1;
  mp.m0w2 = m0w2; mp.m0b2 = m0b2; mp.m0g2 = m0g2; mp.m0t2 = m0t2;
  mp.m0w3 = m0w3; mp.m0b3 = m0b3;
  mp.hw1 = hw1; mp.hb1 = hb1; mp.hg1 = hg1; mp.ht1 = ht1;
  mp.hw2 = hw2; mp.hb2 = hb2; mp.hg2 = hg2; mp.ht2 = ht2;
  mp.hw3 = hw3; mp.hb3 = hb3;
  mp.fcw = fcw; mp.fcb = fcb; mp.lng = lng; mp.lnb = lnb; mp.adot = adot;
  mp.node_dot = node_dot; mp.key_dot = key_dot; mp.attn_out = attn_out;

  mega_kernel<<<N_NODE, 256, 0, stream>>>(mp);

  proj_kernel<<<N_NODE / 16, 256, 0, stream>>>(attn_out, wT, projb, (float*)d_out);
}